// ToRGBLayerSparse_43104291782965
// MI455X (gfx1250) — compile-verified
//
#include <hip/hip_runtime.h>

typedef float v2f __attribute__((ext_vector_type(2)));
typedef float v8f __attribute__((ext_vector_type(8)));

#define IN_CH   256
#define OUT_CH  3
#define W_DIM   512
#define N_BATCH 8

// ---------------------------------------------------------------------------
// Kernel 1 (tiny): styles[b][c] = ((w[b,:] . affine_W[c,:]) * 1/sqrt(512)
//                                  + affine_b[c]) * 1/sqrt(256)
// 2048 outputs, 512-deep dots -> negligible; one block per batch row.
// ---------------------------------------------------------------------------
__global__ void styles_kernel(const float* __restrict__ w,
                              const float* __restrict__ affine_W,
                              const float* __restrict__ affine_b,
                              float* __restrict__ styles) {
  int idx = blockIdx.x * blockDim.x + threadIdx.x;   // 0..2047
  int b = idx >> 8;
  int c = idx & 255;
  const float* wr = w + b * W_DIM;
  const float* ar = affine_W + c * W_DIM;
  float acc = 0.f;
#pragma unroll 8
  for (int k = 0; k < W_DIM; ++k) acc = fmaf(wr[k], ar[k], acc);
  const float affine_gain = 0.04419417382415922f;    // 1/sqrt(512)
  const float weight_gain = 0.0625f;                 // 1/sqrt(256)
  styles[idx] = (acc * affine_gain + affine_b[c]) * weight_gain;
}

// ---------------------------------------------------------------------------
// Kernel 2 (streaming, bandwidth-bound): per wave32, one 16-point tile.
//   A (16x4 f32)  : x_feat rows modulated by styles[batch] (styles in LDS)
//   B (4x16 f32)  : conv_W zero-padded to 16 cols, pre-swizzled per-lane in LDS
//   acc (16x16 f32): 64 chained V_WMMA_F32_16X16X4_F32 over K=256
// Epilogue: lanes with N<3 add bias, clip, store 3 floats per point.
// ---------------------------------------------------------------------------
__global__ void __launch_bounds__(256)
torgb_kernel(const float* __restrict__ x_feat,
             const int*   __restrict__ x_coord,
             const float* __restrict__ styles_g,
             const float* __restrict__ conv_W,
             const float* __restrict__ conv_b,
             float*       __restrict__ out,
             int n_points) {
  __shared__ float s_styles[N_BATCH * IN_CH];  // 8 KB
  __shared__ v2f   s_B[64 * 32];               // 16 KB: per-iter, per-lane B pair

  const int tid = threadIdx.x;

  // Stage styles table.
  for (int t = tid; t < N_BATCH * IN_CH; t += 256) s_styles[t] = styles_g[t];

  // Pre-swizzle B into WMMA lane layout: lane l of iter i needs
  // B[k][n] for n = l&15, k = 4i + 2*(l>>4) + {0,1}; pad n>=3 with 0.
  for (int e = tid; e < 64 * 32; e += 256) {
    int i = e >> 5, l = e & 31;
    int n = l & 15, half = l >> 4;
    int kb = 4 * i + 2 * half;
    v2f v;
    v.x = (n < OUT_CH) ? conv_W[kb * OUT_CH + n]       : 0.f;
    v.y = (n < OUT_CH) ? conv_W[(kb + 1) * OUT_CH + n] : 0.f;
    s_B[e] = v;
  }
  __syncthreads();

  const int wave = tid >> 5;
  const int lane = tid & 31;
  const int tile = blockIdx.x * 8 + wave;
  const int n_tiles = (n_points + 15) >> 4;
  if (tile >= n_tiles) return;   // wave-uniform: EXEC stays all-1s for WMMA

  const int  row  = lane & 15;
  const int  half = lane >> 4;
  long p  = (long)tile * 16 + row;
  long pc = (p < n_points) ? p : (long)(n_points - 1);   // clamp partial tile

  const int    b    = x_coord[pc * 4];          // batch index of this row
  const float* xrow = x_feat + pc * IN_CH;
  const float* srow = s_styles + b * IN_CH;     // LDS

  v8f acc = {};
#pragma unroll 4
  for (int i = 0; i < 64; ++i) {
    int c = 4 * i + 2 * half;
    float2 xv = *(const float2*)(xrow + c);     // global_load_b64, row-sequential
    float2 sv = *(const float2*)(srow + c);     // ds_load_b64
    v2f a;
    a.x = xv.x * sv.x;                          // modulate in registers
    a.y = xv.y * sv.y;
    v2f bb = s_B[i * 32 + lane];                // ds_load_b64, conflict-free
    acc = __builtin_amdgcn_wmma_f32_16x16x4_f32(
        /*neg_a=*/false, a, /*neg_b=*/false, bb,
        /*c_mod=*/(short)0, acc, /*reuse_a=*/false, /*reuse_b=*/false);
  }

  // Epilogue: D layout -> VGPR j, lane l: M = j + 8*(l>>4), N = l&15.
  const int n = lane & 15;
  if (n < OUT_CH) {
    const float bias = conv_b[n];
#pragma unroll
    for (int j = 0; j < 8; ++j) {
      long m = (long)tile * 16 + j + 8 * half;
      if (m < n_points) {
        float v = acc[j] + bias;
        v = fminf(fmaxf(v, -256.f), 256.f);
        out[m * OUT_CH + n] = v;
      }
    }
  }
}

// ---------------------------------------------------------------------------
extern "C" void kernel_launch(void* const* d_in, const int* in_sizes, int n_in,
                              void* d_out, int out_size, void* d_ws, size_t ws_size,
                              hipStream_t stream) {
  const float* x_feat   = (const float*)d_in[0];  // (N, 256) f32
  const int*   x_coord  = (const int*)  d_in[1];  // (N, 4)   i32
  const float* w        = (const float*)d_in[2];  // (8, 512) f32
  const float* affine_W = (const float*)d_in[3];  // (256, 512) f32
  const float* affine_b = (const float*)d_in[4];  // (256,) f32
  const float* conv_W   = (const float*)d_in[5];  // (256, 3) f32
  const float* conv_b   = (const float*)d_in[6];  // (3,) f32
  float*       out      = (float*)d_out;

  const int n_points = in_sizes[0] / IN_CH;
  float* styles = (float*)d_ws;                   // 8 KB scratch

  styles_kernel<<<N_BATCH, 256, 0, stream>>>(w, affine_W, affine_b, styles);

  const int n_tiles = (n_points + 15) / 16;       // 18750 for N=300000
  const int blocks  = (n_tiles + 7) / 8;          // 8 waves (tiles) per block
  torgb_kernel<<<blocks, 256, 0, stream>>>(x_feat, x_coord, styles,
                                           conv_W, conv_b, out, n_points);
}